// BiLSTMCRFTagger_35897336660693
// MI455X (gfx1250) — compile-verified
//
#include <hip/hip_runtime.h>
#include <hip/hip_bf16.h>
#include <math.h>

typedef float v2f __attribute__((ext_vector_type(2)));
typedef float v8f __attribute__((ext_vector_type(8)));

#define VOCAB 50000
#define EMB   256
#define HID   512
#define KTAGS 48
#define HDIM  256           // H = HID/2
#define G4    1024          // 4*H
#define BATCH 64
#define TLEN  512
#define MROWS (BATCH*TLEN)  // 32768

__device__ __forceinline__ float sigm(float v) { return 1.0f / (1.0f + expf(-v)); }

// ---------------------------------------------------------------------------
// Kernel 1: fused embedding-gather + input projection GEMM.
//   Gt[t][b][n] = bias[n] + sum_k emb[x[b*T+t]][k] * Wih[n][k]
// (emb row 0 acts as zeros).  M=32768, N=1024, K=256.
// Each wave: 1 mtile x 4 ntiles -> A fragment reused 4x per k-step.
// Output stored TRANSPOSED to [t][b][4H] so the sequential LSTM scan reads
// one contiguous 256KB block per timestep.
// ---------------------------------------------------------------------------
__global__ __launch_bounds__(256)
void gates_in_gemm(const int* __restrict__ x, const float* __restrict__ emb,
                   const float* __restrict__ Wih, const float* __restrict__ bias,
                   float* __restrict__ Gt) {
  const int tid  = threadIdx.x;
  const int lane = tid & 31;
  const int wave = tid >> 5;
  const int w    = blockIdx.x * 8 + wave;      // 0..32767
  const int mtile = w >> 4;                    // 0..2047
  const int ngrp  = w & 15;                    // 4 consecutive ntiles
  const int m0 = mtile * 16;
  const int ln = lane & 15;
  const bool hi = lane >= 16;
  const int koff = hi ? 2 : 0;

  // A fragment: this lane's matrix row
  const int am = m0 + ln;
  const int xr = x[am];
  const float arow_scale = (xr == 0) ? 0.0f : 1.0f;   // emb.at[0].set(0)
  const float* arow = emb + (size_t)xr * EMB;
  // B fragments: B[k][n] = Wih[n][k]; 4 ntiles, base row ngrp*64 + ln
  const float* bbase = Wih + (size_t)(ngrp * 64 + ln) * EMB;

  v8f acc[4];
#pragma unroll
  for (int q = 0; q < 4; ++q) {
    const float bn = bias[ngrp * 64 + q * 16 + ln];
#pragma unroll
    for (int g = 0; g < 8; ++g) acc[q][g] = bn;
  }

  for (int k = 0; k < EMB; k += 4) {
    v2f a = *(const v2f*)(arow + k + koff);
    a *= arow_scale;
#pragma unroll
    for (int q = 0; q < 4; ++q) {
      v2f b = *(const v2f*)(bbase + (size_t)(q * 16) * EMB + k + koff);
      acc[q] = __builtin_amdgcn_wmma_f32_16x16x4_f32(false, a, false, b,
                                                     (short)0, acc[q], false, false);
    }
  }

  const int crow0 = m0 + (hi ? 8 : 0);
#pragma unroll
  for (int q = 0; q < 4; ++q) {
#pragma unroll
    for (int g = 0; g < 8; ++g) {
      const int r = crow0 + g;            // r = b*T + t  (T=512, tile within one b)
      const int b = r >> 9;
      const int t = r & (TLEN - 1);
      Gt[((size_t)t * BATCH + b) * G4 + (ngrp * 64 + q * 16 + ln)] = acc[q][g];
    }
  }
}

// ---------------------------------------------------------------------------
// Kernel 2: persistent LSTM scan, one workgroup per direction (grid = 2).
// LDS: only h[64][256] (64KB).  Each wave owns the 4 gate quadrants (i,f,g,o)
// for 2 h-column blocks of 16 -> 8 accumulators; the A fragment (one LDS b64
// load per k-step) feeds 8 WMMAs.  Since i/f/g/o land in matching accumulator
// positions, the LSTM cell update runs entirely in registers (c in VGPRs).
// Next timestep's (contiguous) input-gate block is prefetched during GEMM.
// ---------------------------------------------------------------------------
__global__ __launch_bounds__(1024)
void lstm_scan(const float* __restrict__ Gtf, const float* __restrict__ Gtb,
               const float* __restrict__ Whh_f, const float* __restrict__ Whh_b,
               float* __restrict__ hf, float* __restrict__ hb) {
  const int rev = blockIdx.x;                      // 0 = forward, 1 = backward
  const float* __restrict__ G    = rev ? Gtb   : Gtf;
  const float* __restrict__ Whh  = rev ? Whh_b : Whh_f;
  float*       __restrict__ hout = rev ? hb    : hf;

  extern __shared__ float hS[];                    // [64][256]

  const int tid  = threadIdx.x;
  const int lane = tid & 31;
  const int wave = tid >> 5;
  const int ln   = lane & 15;
  const bool hi  = lane >= 16;
  const int koff = hi ? 2 : 0;

  const int m0    = (wave & 3) * 16;
  const int hx0   = (wave >> 2) * 16;              // task adds +128
  const int arow  = m0 + ln;
  const int crow0 = m0 + (hi ? 8 : 0);

  for (int i = tid; i < BATCH * HDIM; i += 1024) hS[i] = 0.0f;

  float c[2][8];
#pragma unroll
  for (int task = 0; task < 2; ++task)
#pragma unroll
    for (int g = 0; g < 8; ++g) c[task][g] = 0.0f;

  // B fragment base: B[k][n] = Whh[n][k]; n = q*256 + task*128 + hx0 + ln
  const float* bbase = Whh + (size_t)(hx0 + ln) * HDIM;

  __syncthreads();

  for (int step = 0; step < TLEN; ++step) {
    const int t = rev ? (TLEN - 1 - step) : step;

    // init accumulators with precomputed input gates (+bias)
    v8f acc[2][4];
    const float* Gstep = G + (size_t)t * BATCH * G4;
#pragma unroll
    for (int task = 0; task < 2; ++task)
#pragma unroll
      for (int q = 0; q < 4; ++q)
#pragma unroll
        for (int g = 0; g < 8; ++g)
          acc[task][q][g] =
              Gstep[(size_t)(crow0 + g) * G4 + (q * HDIM + task * 128 + hx0 + ln)];

    // prefetch next step's contiguous 256KB input-gate block (1024 x 256B)
    if (step + 1 < TLEN) {
      const int tn = rev ? (TLEN - 2 - step) : (step + 1);
      __builtin_prefetch(G + (size_t)tn * BATCH * G4 + (size_t)tid * 64, 0, 1);
    }

    // gates += h @ Whh.T : one LDS A-load -> 8 WMMAs
    for (int k = 0; k < HDIM; k += 4) {
      v2f a = *(const v2f*)(hS + arow * HDIM + k + koff);
#pragma unroll
      for (int task = 0; task < 2; ++task)
#pragma unroll
        for (int q = 0; q < 4; ++q) {
          v2f b = *(const v2f*)(bbase + (size_t)(q * HDIM + task * 128) * HDIM
                                + k + koff);
          acc[task][q] = __builtin_amdgcn_wmma_f32_16x16x4_f32(
              false, a, false, b, (short)0, acc[task][q], false, false);
        }
    }
    __syncthreads();  // all hS reads of this step complete

    // in-register LSTM cell update: acc[.][0..3] = i, f, g, o at same (m,hx)
#pragma unroll
    for (int task = 0; task < 2; ++task) {
      const int hx = hx0 + task * 128 + ln;
#pragma unroll
      for (int g = 0; g < 8; ++g) {
        const float ig = sigm(acc[task][0][g]);
        const float fg = sigm(acc[task][1][g]);
        const float gg = tanhf(acc[task][2][g]);
        const float og = sigm(acc[task][3][g]);
        const float cv = fg * c[task][g] + ig * gg;
        c[task][g] = cv;
        const float hv = og * tanhf(cv);
        const int m = crow0 + g;
        hS[m * HDIM + hx] = hv;
        hout[((size_t)m * TLEN + t) * HDIM + hx] = hv;
      }
    }
    __syncthreads();  // new h visible before next step's GEMM
  }
}

// ---------------------------------------------------------------------------
// Kernel 3: FC to emissions.  emis[r][n] = [hf|hb][r] . fcW[n] + fcb[n]
// M=32768, N=48, K=512.  Each wave: 1 mtile x all 3 ntiles (A reused 3x).
// ---------------------------------------------------------------------------
__global__ __launch_bounds__(256)
void fc_gemm(const float* __restrict__ hf, const float* __restrict__ hb,
             const float* __restrict__ fcW, const float* __restrict__ fcb,
             float* __restrict__ emis) {
  const int tid  = threadIdx.x;
  const int lane = tid & 31;
  const int wave = tid >> 5;
  const int mtile = blockIdx.x * 8 + wave;         // 0..2047
  const int m0 = mtile * 16;
  const int ln = lane & 15;
  const bool hi = lane >= 16;
  const int koff = hi ? 2 : 0;

  const int arow = m0 + ln;
  const float* bbase = fcW + (size_t)ln * HID;     // + q*16 rows

  v8f acc[3];
#pragma unroll
  for (int q = 0; q < 3; ++q) {
    const float bn = fcb[q * 16 + ln];
#pragma unroll
    for (int g = 0; g < 8; ++g) acc[q][g] = bn;
  }

  for (int k = 0; k < HID; k += 4) {
    const float* asrc = (k < HDIM) ? (hf + (size_t)arow * HDIM + k)
                                   : (hb + (size_t)arow * HDIM + (k - HDIM));
    v2f a = *(const v2f*)(asrc + koff);
#pragma unroll
    for (int q = 0; q < 3; ++q) {
      v2f b = *(const v2f*)(bbase + (size_t)(q * 16) * HID + k + koff);
      acc[q] = __builtin_amdgcn_wmma_f32_16x16x4_f32(false, a, false, b,
                                                     (short)0, acc[q], false, false);
    }
  }

  const int crow0 = m0 + (hi ? 8 : 0);
#pragma unroll
  for (int q = 0; q < 3; ++q)
#pragma unroll
    for (int g = 0; g < 8; ++g)
      emis[(size_t)(crow0 + g) * KTAGS + (q * 16 + ln)] = acc[q][g];
}

// ---------------------------------------------------------------------------
// Kernel 4: CRF NLL (single block). Numerator per batch row + forward
// algorithm with double-buffered alpha in LDS, then -mean(num - den).
// ---------------------------------------------------------------------------
__global__ __launch_bounds__(512)
void crf_loss(const float* __restrict__ emis, const int* __restrict__ tags,
              const int* __restrict__ mask, const float* __restrict__ start_t,
              const float* __restrict__ end_t, const float* __restrict__ trans,
              float* __restrict__ out) {
  __shared__ float trS[KTAGS * KTAGS];
  __shared__ float stS[KTAGS], enS[KTAGS];
  __shared__ float alphaA[BATCH * KTAGS], alphaB[BATCH * KTAGS];
  __shared__ float numS[BATCH], partS[BATCH];

  const int tid = threadIdx.x;
  for (int i = tid; i < KTAGS * KTAGS; i += 512) trS[i] = trans[i];
  if (tid < KTAGS) { stS[tid] = start_t[tid]; enS[tid] = end_t[tid]; }
  for (int p = tid; p < BATCH * KTAGS; p += 512) {
    const int b = p / KTAGS, k = p % KTAGS;
    alphaA[p] = start_t[k] + emis[((size_t)b * TLEN) * KTAGS + k];
  }
  __syncthreads();

  // numerator (gold-path score) per batch
  if (tid < BATCH) {
    const int b = tid;
    int tgp = tags[b * TLEN];
    float num = stS[tgp] + emis[((size_t)b * TLEN) * KTAGS + tgp];
    int cnt = (mask[b * TLEN] != 0) ? 1 : 0;
    for (int t = 1; t < TLEN; ++t) {
      const int tg = tags[b * TLEN + t];
      const float mt = (mask[b * TLEN + t] != 0) ? 1.0f : 0.0f;
      num += (trS[tgp * KTAGS + tg] +
              emis[((size_t)b * TLEN + t) * KTAGS + tg]) * mt;
      cnt += (mt > 0.0f) ? 1 : 0;
      tgp = tg;
    }
    const int last = cnt - 1;
    num += enS[tags[b * TLEN + last]];
    numS[b] = num;
  }
  __syncthreads();

  // forward recursion: alpha'[b][j] = em[t][b][j] + lse_i(alpha[b][i]+trans[i][j])
  for (int t = 1; t < TLEN; ++t) {
    const float* cur = (t & 1) ? alphaA : alphaB;
    float*       nxt = (t & 1) ? alphaB : alphaA;
    for (int p = tid; p < BATCH * KTAGS; p += 512) {
      const int b = p / KTAGS, j = p % KTAGS;
      const float* ab = cur + b * KTAGS;
      float mx = -INFINITY;
      for (int i = 0; i < KTAGS; ++i)
        mx = fmaxf(mx, ab[i] + trS[i * KTAGS + j]);
      float s = 0.0f;
      for (int i = 0; i < KTAGS; ++i)
        s += expf(ab[i] + trS[i * KTAGS + j] - mx);
      const float nv = emis[((size_t)b * TLEN + t) * KTAGS + j] + mx + logf(s);
      nxt[p] = (mask[b * TLEN + t] != 0) ? nv : ab[j];
    }
    __syncthreads();
  }

  const float* fin = ((TLEN - 1) & 1) ? alphaB : alphaA;
  if (tid < BATCH) {
    const int b = tid;
    const float* ab = fin + b * KTAGS;
    float mx = -INFINITY;
    for (int k = 0; k < KTAGS; ++k) mx = fmaxf(mx, ab[k] + enS[k]);
    float s = 0.0f;
    for (int k = 0; k < KTAGS; ++k) s += expf(ab[k] + enS[k] - mx);
    partS[b] = numS[b] - (mx + logf(s));
  }
  __syncthreads();
  if (tid == 0) {
    float acc = 0.0f;
    for (int b = 0; b < BATCH; ++b) acc += partS[b];
    out[0] = -acc / (float)BATCH;
  }
}

// ---------------------------------------------------------------------------
extern "C" void kernel_launch(void* const* d_in, const int* in_sizes, int n_in,
                              void* d_out, int out_size, void* d_ws, size_t ws_size,
                              hipStream_t stream) {
  (void)in_sizes; (void)n_in; (void)out_size; (void)ws_size;
  const int*   x       = (const int*)  d_in[0];
  const int*   tags    = (const int*)  d_in[1];
  const int*   mask    = (const int*)  d_in[2];
  const float* emb     = (const float*)d_in[3];
  const float* Wih_f   = (const float*)d_in[4];
  const float* Whh_f   = (const float*)d_in[5];
  const float* b_f     = (const float*)d_in[6];
  const float* Wih_b   = (const float*)d_in[7];
  const float* Whh_b   = (const float*)d_in[8];
  const float* b_b     = (const float*)d_in[9];
  const float* fc_W    = (const float*)d_in[10];
  const float* fc_b    = (const float*)d_in[11];
  const float* start_t = (const float*)d_in[12];
  const float* end_t   = (const float*)d_in[13];
  const float* trans   = (const float*)d_in[14];

  // workspace layout (floats): Gtf, Gtb ([T][B][4H]), hf, hb, emis (~326MiB)
  float* ws   = (float*)d_ws;
  float* Gtf  = ws;
  float* Gtb  = Gtf + (size_t)MROWS * G4;
  float* hf   = Gtb + (size_t)MROWS * G4;
  float* hb   = hf + (size_t)MROWS * HDIM;
  float* emis = hb + (size_t)MROWS * HDIM;

  // 1) input projections (fused embedding gather), both directions
  gates_in_gemm<<<4096, 256, 0, stream>>>(x, emb, Wih_f, b_f, Gtf);
  gates_in_gemm<<<4096, 256, 0, stream>>>(x, emb, Wih_b, b_b, Gtb);

  // 2) recurrent scan, fwd+bwd concurrently (one WGP each, 64KB LDS)
  const size_t ldsBytes = (size_t)BATCH * HDIM * sizeof(float);
  lstm_scan<<<2, 1024, ldsBytes, stream>>>(Gtf, Gtb, Whh_f, Whh_b, hf, hb);

  // 3) emissions
  fc_gemm<<<256, 256, 0, stream>>>(hf, hb, fc_W, fc_b, emis);

  // 4) CRF NLL
  crf_loss<<<1, 512, 0, stream>>>(emis, tags, mask, start_t, end_t, trans,
                                  (float*)d_out);
}